// MyLoss1_44925357916400
// MI455X (gfx1250) — compile-verified
//
#include <hip/hip_runtime.h>

typedef __attribute__((ext_vector_type(4))) float vf4;
typedef __attribute__((ext_vector_type(2))) float v2f;
typedef __attribute__((ext_vector_type(8))) float v8f;

#define BLOCK 256
#define MAXGRID 2048

// Stage 1: each thread streams 4-joint groups (48B from pred, 48B from labels,
// all 16B-aligned b128 non-temporal loads), accumulates in f32, then the wave
// is reduced with V_WMMA_F32_16X16X4_F32 (A = lane accumulators, B = ones),
// and per-block partial sums go to workspace. Fully deterministic (no atomics).
__global__ __launch_bounds__(BLOCK) void loss_partial(const float* __restrict__ pred,
                                                      const float* __restrict__ lab,
                                                      float* __restrict__ ws,
                                                      int units) {
    const int tid    = blockIdx.x * BLOCK + threadIdx.x;
    const int stride = gridDim.x * BLOCK;
    const vf4* p4 = (const vf4*)pred;
    const vf4* l4 = (const vf4*)lab;
    const float third = 1.0f / 3.0f;

    float acc = 0.0f;
    const int iters = (units + stride - 1) / stride;   // uniform trip count -> EXEC full after loop
    for (int k = 0; k < iters; ++k) {
        int i = tid + k * stride;
        if (i < units) {
            long base = (long)i * 3;
            vf4 pa = __builtin_nontemporal_load(&p4[base + 0]);
            vf4 pb = __builtin_nontemporal_load(&p4[base + 1]);
            vf4 pc = __builtin_nontemporal_load(&p4[base + 2]);
            vf4 la = __builtin_nontemporal_load(&l4[base + 0]);
            vf4 lb = __builtin_nontemporal_load(&l4[base + 1]);
            vf4 lc = __builtin_nontemporal_load(&l4[base + 2]);
            float p[12] = {pa.x, pa.y, pa.z, pa.w, pb.x, pb.y, pb.z, pb.w, pc.x, pc.y, pc.z, pc.w};
            float q[12] = {la.x, la.y, la.z, la.w, lb.x, lb.y, lb.z, lb.w, lc.x, lc.y, lc.z, lc.w};
#pragma unroll
            for (int j = 0; j < 4; ++j) {
                float d0 = p[3*j + 0] - q[3*j + 0];
                float d1 = p[3*j + 1] - q[3*j + 1];
                float d2 = p[3*j + 2] - q[3*j + 2];
                float s0 = d0 * d0, s1 = d1 * d1, s2 = d2 * d2;
                float full = (s0 + s1 + s2) * third;
                acc += (p[3*j + 2] > 0.5f) ? full : s2;
            }
        }
    }

    // Wave32 reduction via WMMA: A(16x4) has acc in K=0 (lanes 0-15) / K=2 (lanes 16-31),
    // zeros elsewhere; B = ones(4x16)  =>  D[m,n] = acc_m + acc_{m+16} for every column n.
    v2f a; a[0] = acc;  a[1] = 0.0f;
    v2f b; b[0] = 1.0f; b[1] = 1.0f;
    v8f c = {};
    c = __builtin_amdgcn_wmma_f32_16x16x4_f32(false, a, false, b, (short)0, c, false, false);
    // Lane L<16 holds rows 0-7 of column L; lane L+16 holds rows 8-15 of column L.
    float rowsum = ((c[0] + c[1]) + (c[2] + c[3])) + ((c[4] + c[5]) + (c[6] + c[7]));
    float wsum = rowsum + __shfl_xor(rowsum, 16, 32);   // full 32-lane sum, exact f32 order

    __shared__ float smem[BLOCK / 32];
    const int lane = threadIdx.x & 31;
    const int wave = threadIdx.x >> 5;
    if (lane == 0) smem[wave] = wsum;
    __syncthreads();
    if (threadIdx.x == 0) {
        float s = 0.0f;
#pragma unroll
        for (int w = 0; w < BLOCK / 32; ++w) s += smem[w];
        ws[blockIdx.x] = s;
    }
}

// Stage 2: single block, fixed-order reduction of per-block partials + scalar
// tail joints (joints % 4), scale by 1/B, overwrite d_out[0].
__global__ __launch_bounds__(BLOCK) void loss_final(const float* __restrict__ ws, int nblocks,
                                                    const float* __restrict__ pred,
                                                    const float* __restrict__ lab,
                                                    int tail_start, int joints_total,
                                                    float inv_b, float* __restrict__ out) {
    __shared__ float smem[BLOCK];
    float s = 0.0f;
    for (int i = threadIdx.x; i < nblocks; i += BLOCK) s += ws[i];   // fixed order
    smem[threadIdx.x] = s;
    __syncthreads();
#pragma unroll
    for (int off = BLOCK / 2; off > 0; off >>= 1) {
        if ((int)threadIdx.x < off) smem[threadIdx.x] += smem[threadIdx.x + off];
        __syncthreads();
    }
    if (threadIdx.x == 0) {
        float total = smem[0];
        for (int j = tail_start; j < joints_total; ++j) {   // empty for B even
            float d0 = pred[3*j + 0] - lab[3*j + 0];
            float d1 = pred[3*j + 1] - lab[3*j + 1];
            float d2 = pred[3*j + 2] - lab[3*j + 2];
            float s2 = d2 * d2;
            float full = (d0 * d0 + d1 * d1 + s2) * (1.0f / 3.0f);
            total += (pred[3*j + 2] > 0.5f) ? full : s2;
        }
        out[0] = total * inv_b;
    }
}

extern "C" void kernel_launch(void* const* d_in, const int* in_sizes, int n_in,
                              void* d_out, int out_size, void* d_ws, size_t ws_size,
                              hipStream_t stream) {
    const float* pred = (const float*)d_in[0];
    const float* lab  = (const float*)d_in[1];
    float* out = (float*)d_out;
    float* ws  = (float*)d_ws;

    const long n      = (long)in_sizes[0];  // B * 54
    const long joints = n / 3;              // B * 18
    const int  units  = (int)(joints / 4);  // 4-joint groups (48B per array)
    const int  tail   = units * 4;
    const long Bn     = n / 54;

    int grid = MAXGRID;
    if ((size_t)grid * sizeof(float) > ws_size) grid = (int)(ws_size / sizeof(float));
    if (grid < 1) grid = 1;

    loss_partial<<<grid, BLOCK, 0, stream>>>(pred, lab, ws, units);
    loss_final<<<1, BLOCK, 0, stream>>>(ws, grid, pred, lab, tail, (int)joints,
                                        1.0f / (float)Bn, out);
}